// APPNP_20993800142878
// MI455X (gfx1250) — compile-verified
//
#include <hip/hip_runtime.h>

typedef __attribute__((ext_vector_type(2))) float v2f;
typedef __attribute__((ext_vector_type(8))) float v8f;

#define N_NODES   50000
#define N_EDGES   1600000
#define IN_C      512
#define HID_C     256
#define OUT_C     50
#define OUT_P     64      // padded channel count (16-aligned, 2x wave32)
#define ALPHA     0.1f
#define NUM_LAYERS 10

__device__ __forceinline__ v8f wmma_f32(v2f a, v2f b, v8f c) {
    // D = A(16x4 f32) * B(4x16 f32) + C(16x16 f32), exact fp32 matrix op
    return __builtin_amdgcn_wmma_f32_16x16x4_f32(
        /*neg_a=*/false, a, /*neg_b=*/false, b,
        /*c_mod=*/(short)0, c, /*reuse_a=*/false, /*reuse_b=*/false);
}

// Force the native GLOBAL_ATOMIC_ADD_F32 (non-returning, STOREcnt-tracked)
// instead of a possible CAS loop from plain atomicAdd.
__device__ __forceinline__ void atomic_add_f32(float* p, float v) {
    unsafeAtomicAdd(p, v);
}

// ---------------------------------------------------------------------------
// GEMM1: h1 = relu(x[50000,512] @ W1[512,256] + b1)   (per-wave 32x32 tile)
// ---------------------------------------------------------------------------
__global__ __launch_bounds__(256) void gemm1_relu(const float* __restrict__ x,
                                                  const float* __restrict__ W1,
                                                  const float* __restrict__ b1,
                                                  float* __restrict__ h1) {
    const int lane = threadIdx.x & 31;
    const int wave = threadIdx.x >> 5;
    const int lm   = lane & 15;
    const int half = lane >> 4;

    const int w   = blockIdx.x * 8 + wave;   // 0..12503 wave-tiles
    const int m32 = w >> 3;                  // 1563 row strips of 32
    const int n32 = w & 7;                   // 8 col strips of 32
    const int row0 = m32 * 32;
    const int col0 = n32 * 32;

    v8f acc00 = {}, acc01 = {}, acc10 = {}, acc11 = {};

    int r0 = row0 + lm;
    int r1 = row0 + 16 + lm;
    if (r0 > N_NODES - 1) r0 = N_NODES - 1;   // clamp loads; stores guarded
    if (r1 > N_NODES - 1) r1 = N_NODES - 1;

    const float* a0p = x + (size_t)r0 * IN_C + 2 * half;
    const float* a1p = x + (size_t)r1 * IN_C + 2 * half;
    // B frag: component i = W1[(k + i + 2*half)][col + lm]
    const float* bp  = W1 + (size_t)(2 * half) * HID_C + col0 + lm;

    for (int k = 0; k < IN_C; k += 4) {
        v2f a0 = *(const v2f*)(a0p + k);
        v2f a1 = *(const v2f*)(a1p + k);
        const float* bk = bp + (size_t)k * HID_C;
        v2f b0, b1v;
        b0.x  = bk[0];        b0.y  = bk[HID_C];
        b1v.x = bk[16];       b1v.y = bk[HID_C + 16];
        acc00 = wmma_f32(a0, b0,  acc00);
        acc01 = wmma_f32(a0, b1v, acc01);
        acc10 = wmma_f32(a1, b0,  acc10);
        acc11 = wmma_f32(a1, b1v, acc11);
    }

    const float bias0 = b1[col0 + lm];
    const float bias1 = b1[col0 + 16 + lm];
    #pragma unroll
    for (int i = 0; i < 8; ++i) {
        const int rA = row0 + i + 8 * half;        // rows of acc0x
        const int rB = rA + 16;                    // rows of acc1x
        if (rA < N_NODES) {
            float v0 = acc00[i] + bias0;
            float v1 = acc01[i] + bias1;
            h1[(size_t)rA * HID_C + col0 + lm]      = v0 > 0.0f ? v0 : 0.0f;
            h1[(size_t)rA * HID_C + col0 + 16 + lm] = v1 > 0.0f ? v1 : 0.0f;
        }
        if (rB < N_NODES) {
            float v0 = acc10[i] + bias0;
            float v1 = acc11[i] + bias1;
            h1[(size_t)rB * HID_C + col0 + lm]      = v0 > 0.0f ? v0 : 0.0f;
            h1[(size_t)rB * HID_C + col0 + 16 + lm] = v1 > 0.0f ? v1 : 0.0f;
        }
    }
}

// ---------------------------------------------------------------------------
// GEMM2: x0p = h1[50000,256] @ W2[256,50] + b2, stored 64-padded (pads = 0)
// ---------------------------------------------------------------------------
__global__ __launch_bounds__(256) void gemm2(const float* __restrict__ h1,
                                             const float* __restrict__ W2,
                                             const float* __restrict__ b2,
                                             float* __restrict__ x0p) {
    const int lane = threadIdx.x & 31;
    const int wave = threadIdx.x >> 5;
    const int lm   = lane & 15;
    const int half = lane >> 4;

    const int tile = blockIdx.x * 8 + wave;     // 16-row tiles, 3125 total
    if (tile >= N_NODES / 16) return;           // wave-uniform guard
    const int row0 = tile * 16;

    v8f acc[4] = {{}, {}, {}, {}};
    const float* ap = h1 + (size_t)(row0 + lm) * HID_C + 2 * half;

    for (int k = 0; k < HID_C; k += 4) {
        v2f a = *(const v2f*)(ap + k);
        const int kr = k + 2 * half;
        #pragma unroll
        for (int t = 0; t < 4; ++t) {
            const int n = t * 16 + lm;
            v2f b;
            b.x = (n < OUT_C) ? W2[(size_t)kr * OUT_C + n]       : 0.0f;
            b.y = (n < OUT_C) ? W2[(size_t)(kr + 1) * OUT_C + n] : 0.0f;
            acc[t] = wmma_f32(a, b, acc[t]);
        }
    }

    #pragma unroll
    for (int t = 0; t < 4; ++t) {
        const int n = t * 16 + lm;
        const float bias = (n < OUT_C) ? b2[n] : 0.0f;
        #pragma unroll
        for (int i = 0; i < 8; ++i) {
            const int r = row0 + i + 8 * half;
            const float v = (n < OUT_C) ? (acc[t][i] + bias) : 0.0f;
            x0p[(size_t)r * OUT_P + n] = v;
        }
    }
}

// ---------------------------------------------------------------------------
// Propagation: h_next = alpha * x0   (then scatter adds 0.9*w*h_cur[col])
// ---------------------------------------------------------------------------
__global__ __launch_bounds__(256) void init_next(const float* __restrict__ x0p,
                                                 float* __restrict__ nxt, int n) {
    const int i = blockIdx.x * blockDim.x + threadIdx.x;
    if (i < n) nxt[i] = ALPHA * x0p[i];
}

__global__ __launch_bounds__(256) void scatter(const int* __restrict__ erow,
                                               const int* __restrict__ ecol,
                                               const float* __restrict__ ewgt,
                                               const float* __restrict__ cur,
                                               float* __restrict__ nxt) {
    const int e = (blockIdx.x * blockDim.x + threadIdx.x) >> 5;  // wave/edge
    const int c = threadIdx.x & 31;
    if (e >= N_EDGES) return;
    const int   r  = erow[e];
    const int   cl = ecol[e];
    const float w  = (1.0f - ALPHA) * ewgt[e];
    const float* src = cur + (size_t)cl * OUT_P;
    float*       dst = nxt + (size_t)r  * OUT_P;
    if (c < OUT_C)       atomic_add_f32(dst + c,  w * src[c]);
    const int c2 = c + 32;
    if (c2 < OUT_C)      atomic_add_f32(dst + c2, w * src[c2]);
}

// ---------------------------------------------------------------------------
// log_softmax over 50 classes, one wave32 per node
// ---------------------------------------------------------------------------
__global__ __launch_bounds__(256) void logsoftmax_k(const float* __restrict__ h,
                                                    float* __restrict__ out) {
    const int node = (blockIdx.x * blockDim.x + threadIdx.x) >> 5;
    const int c = threadIdx.x & 31;
    if (node >= N_NODES) return;
    const float* hp = h + (size_t)node * OUT_P;
    const float NEG_INF = -__builtin_inff();
    const int c2 = c + 32;
    float v0 = (c  < OUT_C) ? hp[c]  : NEG_INF;
    float v1 = (c2 < OUT_C) ? hp[c2] : NEG_INF;

    float m = fmaxf(v0, v1);
    #pragma unroll
    for (int off = 16; off > 0; off >>= 1)
        m = fmaxf(m, __shfl_xor(m, off, 32));

    float s = 0.0f;
    if (c  < OUT_C) s += expf(v0 - m);
    if (c2 < OUT_C) s += expf(v1 - m);
    #pragma unroll
    for (int off = 16; off > 0; off >>= 1)
        s += __shfl_xor(s, off, 32);

    const float l = m + logf(s);
    float* op = out + (size_t)node * OUT_C;
    if (c  < OUT_C) op[c]  = v0 - l;
    if (c2 < OUT_C) op[c2] = v1 - l;
}

// ---------------------------------------------------------------------------
extern "C" void kernel_launch(void* const* d_in, const int* in_sizes, int n_in,
                              void* d_out, int out_size, void* d_ws, size_t ws_size,
                              hipStream_t stream) {
    const float* x    = (const float*)d_in[0];
    const int*   erow = (const int*)  d_in[1];
    const int*   ecol = (const int*)  d_in[2];
    const float* ew   = (const float*)d_in[3];
    const float* W1   = (const float*)d_in[4];
    const float* b1   = (const float*)d_in[5];
    const float* W2   = (const float*)d_in[6];
    const float* b2   = (const float*)d_in[7];
    float* out = (float*)d_out;

    float* h1  = (float*)d_ws;                        // 50000*256 f32
    float* x0p = h1  + (size_t)N_NODES * HID_C;       // 50000*64 f32
    float* hA  = x0p + (size_t)N_NODES * OUT_P;       // 50000*64 f32
    float* hB  = hA  + (size_t)N_NODES * OUT_P;       // 50000*64 f32

    // MLP (fp32 WMMA)
    gemm1_relu<<<1563, 256, 0, stream>>>(x, W1, b1, h1);
    gemm2<<<(N_NODES / 16 + 7) / 8, 256, 0, stream>>>(h1, W2, b2, x0p);

    // 10 propagation rounds: h = 0.9*SpMM(h) + 0.1*x0
    const int nelem = N_NODES * OUT_P;
    const int initBlocks    = (nelem + 255) / 256;
    const int scatterBlocks = (N_EDGES + 7) / 8;      // wave per edge, 8/block
    const float* cur = x0p;
    float* bufs[2] = {hA, hB};
    for (int t = 0; t < NUM_LAYERS; ++t) {
        float* nxt = bufs[t & 1];
        init_next<<<initBlocks, 256, 0, stream>>>(x0p, nxt, nelem);
        scatter<<<scatterBlocks, 256, 0, stream>>>(erow, ecol, ew, cur, nxt);
        cur = nxt;
    }

    // log_softmax -> dense [50000, 50] output
    logsoftmax_k<<<(N_NODES + 7) / 8, 256, 0, stream>>>(cur, out);
    (void)in_sizes; (void)n_in; (void)out_size; (void)ws_size;
}